// InteractionArch_72198400245838
// MI455X (gfx1250) — compile-verified
//
#include <hip/hip_runtime.h>

typedef __attribute__((ext_vector_type(2))) float v2f;
typedef __attribute__((ext_vector_type(8))) float v8f;

#define NUM_SPARSE 26
#define NFEAT 27                   // 1 dense + 26 sparse
#define DIM 128
#define TRI (NFEAT * (NFEAT + 1) / 2)   // 378
#define ROWSTRIDE 132              // 128 + 4: stride%64==4 -> conflict-free b64 frag loads
#define WAVES_PER_BLOCK 4
#define SLAB (NFEAT * ROWSTRIDE)   // 3564 floats per wave (>= 1024 needed for C staging)

__global__ __launch_bounds__(WAVES_PER_BLOCK * 32)
void dlrm_interaction_wmma(const float* __restrict__ dense,
                           const float* __restrict__ sparse,
                           float* __restrict__ out,
                           int B)
{
    __shared__ __align__(16) float slabs[WAVES_PER_BLOCK][SLAB];
    __shared__ __align__(16) float zerorow[DIM];

    const int tid  = threadIdx.x;
    const int wave = tid >> 5;
    const int lane = tid & 31;
    const int b    = blockIdx.x * WAVES_PER_BLOCK + wave;
    const bool active = (b < B);

    float* slab = slabs[wave];

    // shared zero row (used to pad Gram rows 27..31)
    zerorow[tid & (DIM - 1)] = 0.0f;

    // ---- stage X[27 x 128] (fp32) for this wave's batch element into LDS ----
    if (active) {
        const float* drow = dense + (size_t)b * DIM;
        float4 v0 = *(const float4*)(drow + lane * 4);
        *(float4*)(slab + 0 * ROWSTRIDE + lane * 4) = v0;
        const float* srow = sparse + (size_t)b * (NUM_SPARSE * DIM);
        #pragma unroll
        for (int r = 1; r < NFEAT; ++r) {
            float4 s = *(const float4*)(srow + (size_t)(r - 1) * DIM + lane * 4);
            *(float4*)(slab + r * ROWSTRIDE + lane * 4) = s;
        }
    }
    __syncthreads();

    // ---- G = X * X^T via V_WMMA_F32_16X16X4_F32, 2x2 tile grid, triu tiles only ----
    // 32-bit A 16x4 layout: lanes 0-15 hold K=0(K=1 in vgpr1), lanes 16-31 hold K=2,3.
    // For a Gram matrix, B-fragment(col tile j) == A-fragment(row tile j).
    const int n   = lane & 15;
    const int sub = (lane >> 4) << 1;   // 0 or 2

    const float* baseA = slab + n * ROWSTRIDE;                        // rows 0..15
    const float* baseB = (16 + n < NFEAT) ? (slab + (16 + n) * ROWSTRIDE)
                                          : zerorow;                  // rows 16..31 (27..31 -> 0)

    v8f acc00 = {};
    v8f acc01 = {};
    v8f acc11 = {};

    #pragma unroll
    for (int kk = 0; kk < DIM / 4; ++kk) {
        const int col = kk * 4 + sub;
        v2f fa = *(const v2f*)(baseA + col);
        v2f fb = *(const v2f*)(baseB + col);
        acc00 = __builtin_amdgcn_wmma_f32_16x16x4_f32(false, fa, false, fa,
                                                      (short)0, acc00, false, false);
        acc01 = __builtin_amdgcn_wmma_f32_16x16x4_f32(false, fa, false, fb,
                                                      (short)0, acc01, false, false);
        acc11 = __builtin_amdgcn_wmma_f32_16x16x4_f32(false, fb, false, fb,
                                                      (short)0, acc11, false, false);
    }

    // ---- stage C (32x32) into LDS (reuse slab; X no longer needed) ----
    // C/D layout: lanes 0-15 -> M=v, N=lane; lanes 16-31 -> M=v+8, N=lane-16.
    {
        const int mbase = (lane < 16) ? 0 : 8;
        #pragma unroll
        for (int v = 0; v < 8; ++v) {
            const int m = mbase + v;
            slab[m * 32 + n]              = acc00[v];
            slab[m * 32 + 16 + n]         = acc01[v];
            slab[(16 + m) * 32 + 16 + n]  = acc11[v];
        }
    }
    __syncthreads();

    // ---- pack upper triangle (incl. diagonal), row-major triu order, coalesced ----
    if (active) {
        float* outb = out + (size_t)b * TRI;
        int i = 0, base = 0;
        for (int t = lane; t < TRI; t += 32) {
            while (t >= base + (NFEAT - i)) { base += NFEAT - i; ++i; }
            const int j = i + (t - base);
            outb[t] = slab[i * 32 + j];
        }
    }
}

extern "C" void kernel_launch(void* const* d_in, const int* in_sizes, int n_in,
                              void* d_out, int out_size, void* d_ws, size_t ws_size,
                              hipStream_t stream) {
    const float* dense  = (const float*)d_in[0];   // [B, 128] fp32
    const float* sparse = (const float*)d_in[1];   // [B, 26*128] fp32
    float* out = (float*)d_out;                    // [B, 378] fp32

    const int B = in_sizes[0] / DIM;
    const int blocks = (B + WAVES_PER_BLOCK - 1) / WAVES_PER_BLOCK;

    hipLaunchKernelGGL(dlrm_interaction_wmma,
                       dim3(blocks), dim3(WAVES_PER_BLOCK * 32), 0, stream,
                       dense, sparse, out, B);
}